// STFT_88338887344336
// MI455X (gfx1250) — compile-verified
//
#include <hip/hip_runtime.h>
#include <hip/hip_bf16.h>

// ---------------- problem constants (from the reference) ----------------
#define BATCH      64
#define T_WAV      160000
#define N_FFT      512
#define HOP        256
#define PAD_LEFT   256
#define N_FREQ     257          // cos rows; sin rows follow -> 514 kernel rows
#define K_ROWS     514
#define N_FRAMES   625          // floor((160511-512)/256)+1
#define FRAME_TILE 16
#define FREQ_WAVE  16           // freq rows per wave
#define WAVES_PER_BLOCK 4
#define FREQ_TILE  (FREQ_WAVE * WAVES_PER_BLOCK)   // 64
#define N_FRAME_TILES 40        // ceil(625/16)
#define N_FREQ_TILES  9         // ceil(514/64)
#define X_TILE_ELEMS  (FRAME_TILE * HOP + N_FFT)   // 4608 floats = 18 KB
#define SPEC_ELEMS    ((size_t)BATCH * N_FREQ * N_FRAMES * 2)  // 20,560,000
#define INV_WIN_SUM   (1.0f / 256.0f)

typedef float v2f __attribute__((ext_vector_type(2)));
typedef float v8f __attribute__((ext_vector_type(8)));

// D(16 freq rows x 16 frames) = A(16 freq rows x 512) * B(512 x 16 frames)
// A fragment: straight from global (each element consumed exactly once per wave;
//             cross-block reuse handled by L2 where the 1 MB DFT kernel is hot).
// B fragment: frames staged in LDS, shared by the 4 waves of the block.
__global__ __launch_bounds__(128) void stft_wmma_kernel(
    const float* __restrict__ wav,      // [64, 160000]
    const float* __restrict__ dftk,     // [514, 512] (squeezed [514,1,512])
    float* __restrict__ out)            // [64, 257, 625, 2]
{
    __shared__ float xs[X_TILE_ELEMS];

    const int f0 = blockIdx.x * FRAME_TILE;            // first frame of tile
    const int rg = blockIdx.y * FREQ_TILE;             // first kernel row of block
    const int b  = blockIdx.z;                         // batch

    // ---- stage the 16-frame sample window into LDS (handles center padding) ----
    const float* wavb = wav + (size_t)b * T_WAV;
    const int sbase = f0 * HOP - PAD_LEFT;             // wav-space index of xs[0]
    for (int i = threadIdx.x; i < X_TILE_ELEMS; i += 128) {
        const int s = sbase + i;
        xs[i] = (s >= 0 && s < T_WAV) ? wavb[s] : 0.0f;
    }
    __syncthreads();

    const int lane = threadIdx.x & 31;
    const int wave = threadIdx.x >> 5;
    const int half = lane >> 4;                        // 0: lanes 0-15, 1: lanes 16-31
    const int l16  = lane & 15;
    const int koff = half << 1;                        // K sub-offset {0,2} per ISA layout

    // A-matrix source: kernel row per lane (clamped for padded rows; stores masked later)
    const int r_base = rg + wave * FREQ_WAVE;
    int arow = r_base + l16;
    if (arow > K_ROWS - 1) arow = K_ROWS - 1;
    const float* aptr = dftk + (size_t)arow * N_FFT + koff;

    // B-matrix source: frame l16's samples in LDS
    const float* bptr = xs + l16 * HOP + koff;

    v8f acc = {};
#pragma unroll 8
    for (int kk = 0; kk < N_FFT; kk += 4) {
        v2f a = *(const v2f*)(aptr + kk);              // global_load_b64 (L2-resident)
        v2f bfrag = *(const v2f*)(bptr + kk);          // ds_load_b64
        acc = __builtin_amdgcn_wmma_f32_16x16x4_f32(
            /*neg_a=*/false, a, /*neg_b=*/false, bfrag,
            /*c_mod=*/(short)0, acc, /*reuse_a=*/false, /*reuse_b=*/false);
    }

    // ---- scatter D: lane holds D[M=v(+8 for upper half)][N=l16] in acc[v] ----
    const int frame = f0 + l16;
    if (frame < N_FRAMES) {
#pragma unroll
        for (int v = 0; v < 8; ++v) {
            const int r = r_base + v + (half << 3);    // kernel row (freq index incl. sin half)
            if (r < K_ROWS) {
                const int kf = (r < N_FREQ) ? r : (r - N_FREQ);
                const int c  = (r < N_FREQ) ? 0 : 1;
                const size_t idx =
                    ((((size_t)b * N_FREQ + kf) * N_FRAMES) + frame) * 2 + c;
                out[idx] = acc[v] * INV_WIN_SUM;
            }
        }
    }
}

// spec_len = floor((wav_len - 1)/256) + 1  (matches the reference float formula)
__global__ void spec_len_kernel(const int* __restrict__ wav_len,
                                int* __restrict__ len_out)
{
    const int i = threadIdx.x;
    if (i < BATCH) {
        const int wl = wav_len[i];
        len_out[i] = (wl <= 0) ? 0 : ((wl - 1) / HOP + 1);
    }
}

extern "C" void kernel_launch(void* const* d_in, const int* in_sizes, int n_in,
                              void* d_out, int out_size, void* d_ws, size_t ws_size,
                              hipStream_t stream) {
    const float* wav     = (const float*)d_in[0];   // [64,160000] f32
    const float* dftk    = (const float*)d_in[1];   // [514,1,512] f32
    const int*   wav_len = (const int*)d_in[2];     // [64] i32
    float* out = (float*)d_out;

    dim3 grid(N_FRAME_TILES, N_FREQ_TILES, BATCH);  // 40 x 9 x 64
    stft_wmma_kernel<<<grid, dim3(128), 0, stream>>>(wav, dftk, out);

    // spec_len stored as raw int32 bits after the 20,560,000 spec floats
    spec_len_kernel<<<1, BATCH, 0, stream>>>(wav_len, (int*)(out + SPEC_ELEMS));
}